// ScorePredictor_61495341744685
// MI455X (gfx1250) — compile-verified
//
#include <hip/hip_runtime.h>

// CDNA5 / gfx1250: wave32, WMMA 16x16x4 f32.
typedef __attribute__((ext_vector_type(2))) float v2f;
typedef __attribute__((ext_vector_type(8))) float v8f;

#define D_FEAT 64
#define WAVES_PER_BLOCK 8   // 256 threads = 8 wave32

__device__ __forceinline__ float dot4(float4 a, float4 b) {
    return fmaf(a.x, b.x, fmaf(a.y, b.y, fmaf(a.z, b.z, a.w * b.w)));
}

// One wave handles 16 edges per tile.
// Lane n and lane n+16 co-own edge n: lane n covers K=0..31, lane n+16 covers K=32..63.
// Each lane produces 2 partial sums -> exactly the B-matrix (4x16 f32, 2 VGPRs/lane)
// layout of V_WMMA_F32_16X16X4_F32. With A = all-ones, D[m][n] = sum of column n,
// i.e. one WMMA completes all 16 edge dot products in full f32.
__global__ __launch_bounds__(256) void edge_dot_wmma_kernel(
    const float* __restrict__ h,
    const int*   __restrict__ srcIdx,
    const int*   __restrict__ dstIdx,
    float*       __restrict__ out,
    int numTiles)
{
    const int lane = threadIdx.x & 31;
    const int n    = lane & 15;   // edge slot within tile
    const int half = lane >> 4;   // 0: K 0..31, 1: K 32..63

    const int wave     = (blockIdx.x * blockDim.x + threadIdx.x) >> 5; // uniform per wave
    const int numWaves = (gridDim.x * blockDim.x) >> 5;

    const v2f aOnes = {1.0f, 1.0f};   // A = 16x4 matrix of ones (any layout: all ones)

    for (int tile = wave; tile < numTiles; tile += numWaves) {
        const int e = tile * 16 + n;

        // Coalesced 64B index loads per half-wave (halves read identical addrs -> L0 hit).
        const int srow = srcIdx[e];
        const int drow = dstIdx[e];

        const float4* __restrict__ up =
            (const float4*)(h + (long)srow * D_FEAT + half * 32);
        const float4* __restrict__ vp =
            (const float4*)(h + (long)drow * D_FEAT + half * 32);

        // 16 x global_load_b128 per lane: each 256B feature row is read exactly once.
        float4 u[8], w[8];
#pragma unroll
        for (int i = 0; i < 8; ++i) u[i] = up[i];
#pragma unroll
        for (int i = 0; i < 8; ++i) w[i] = vp[i];

        // Two f32 partials per lane (16 products each) -> B matrix registers.
        float p0 = dot4(u[0], w[0]);
        p0 += dot4(u[1], w[1]);
        p0 += dot4(u[2], w[2]);
        p0 += dot4(u[3], w[3]);
        float p1 = dot4(u[4], w[4]);
        p1 += dot4(u[5], w[5]);
        p1 += dot4(u[6], w[6]);
        p1 += dot4(u[7], w[7]);

        v2f b;
        b.x = p0;
        b.y = p1;

        v8f c = {};
        // D = A(ones) x B + 0  ==> every row of D = per-column sums = the 16 edge scores.
        // 8 args: (neg_a, A, neg_b, B, c_mod, C, reuse_a, reuse_b)
        v8f d = __builtin_amdgcn_wmma_f32_16x16x4_f32(
            false, aOnes, false, b, (short)0, c, false, false);

        // D VGPR0: lanes 0-15 hold M=0, N=lane -> the score for edge slot n.
        // Coalesced 64B store from the low half-wave.
        if (half == 0) {
            out[tile * 16 + n] = d[0];
        }
    }
}

// Scalar fallback for E % 16 edges (not hit for E = 1,000,000; kept for generality).
__global__ void edge_dot_tail_kernel(
    const float* __restrict__ h,
    const int*   __restrict__ srcIdx,
    const int*   __restrict__ dstIdx,
    float*       __restrict__ out,
    int start, int E)
{
    int e = start + blockIdx.x * blockDim.x + threadIdx.x;
    if (e >= E) return;
    const float* u = h + (long)srcIdx[e] * D_FEAT;
    const float* v = h + (long)dstIdx[e] * D_FEAT;
    float s = 0.0f;
#pragma unroll
    for (int k = 0; k < D_FEAT; ++k) s = fmaf(u[k], v[k], s);
    out[e] = s;
}

extern "C" void kernel_launch(void* const* d_in, const int* in_sizes, int n_in,
                              void* d_out, int out_size, void* d_ws, size_t ws_size,
                              hipStream_t stream) {
    const float* h      = (const float*)d_in[0];
    const int*   srcIdx = (const int*)d_in[1];
    const int*   dstIdx = (const int*)d_in[2];
    float*       out    = (float*)d_out;

    const int E        = in_sizes[1];   // number of edges
    const int numTiles = E / 16;        // 16 edges per wave-tile

    if (numTiles > 0) {
        // One wave per tile (grid-stride loop handles any mismatch).
        int blocks = (numTiles + WAVES_PER_BLOCK - 1) / WAVES_PER_BLOCK;
        edge_dot_wmma_kernel<<<blocks, 256, 0, stream>>>(h, srcIdx, dstIdx, out, numTiles);
    }

    const int rem = E - numTiles * 16;
    if (rem > 0) {
        edge_dot_tail_kernel<<<1, 64, 0, stream>>>(h, srcIdx, dstIdx, out, numTiles * 16, E);
    }
}